// ChargeEquilibration_8134668059078
// MI455X (gfx1250) — compile-verified
//
#include <hip/hip_runtime.h>
#include <math.h>

typedef __attribute__((ext_vector_type(2))) float v2f;
typedef __attribute__((ext_vector_type(8))) float v8f;

#define MM 64
#define BOHR 0.5291772109f
#define SQRTPI 1.7724538509055160f

__global__ __launch_bounds__(256)
void qeq_fused_kernel(const int* __restrict__ index,
                      const float* __restrict__ positions,
                      const float* __restrict__ hardness,
                      const float* __restrict__ chi,
                      const float* __restrict__ sigma,
                      const float* __restrict__ total_charge,
                      float* __restrict__ out_charge,
                      float* __restrict__ out_energy)
{
    __shared__ float F[MM][MM];   // factor matrix, Cholesky L in lower triangle
    __shared__ float P[MM][MM];   // pairwise pv (zero diag, zero invalid)
    __shared__ float pos[MM][3];
    __shared__ float sig[MM], hrd[MM], eln[MM];
    __shared__ float rhs[2][MM];  // z then y, in place
    __shared__ float red[256];
    __shared__ int   nvalid;
    __shared__ float lambda_s;

    const int b   = blockIdx.x;
    const int tid = threadIdx.x;

    // ---------------- load per-atom data ----------------
    if (tid < MM) {
        const int i = tid;
        pos[i][0] = positions[(b*MM + i)*3 + 0];
        pos[i][1] = positions[(b*MM + i)*3 + 1];
        pos[i][2] = positions[(b*MM + i)*3 + 2];
        sig[i] = sigma[b*MM + i];
        hrd[i] = hardness[b*MM + i];
        eln[i] = chi[b*MM + i];
        red[i] = (index[b*MM + i] != -1) ? 1.0f : 0.0f;
    }
    __syncthreads();
    if (tid == 0) {
        int cnt = 0;
        for (int i = 0; i < MM; ++i) cnt += (red[i] > 0.5f) ? 1 : 0;
        nvalid = cnt;
    }
    __syncthreads();
    const int n = nvalid;

    // ---------------- build P and F ----------------
    for (int e = tid; e < MM*MM; e += 256) {
        const int i = e >> 6, j = e & 63;
        const bool vi = (i < n), vj = (j < n);
        float pv = 0.0f, f;
        if (vi && vj && (i != j)) {
            const float dx = pos[i][0]-pos[j][0];
            const float dy = pos[i][1]-pos[j][1];
            const float dz = pos[i][2]-pos[j][2];
            const float d  = sqrtf(dx*dx + dy*dy + dz*dz);
            const float g  = rsqrtf(2.0f*(sig[i]*sig[i] + sig[j]*sig[j]));
            pv = erff(d*g)/d * BOHR;
            f  = pv;
        } else if (i == j) {
            f = vi ? (hrd[i] + 1.0f/(sig[i]*SQRTPI))*BOHR : 1.0f;
        } else {
            f = 0.0f;
        }
        P[i][j] = pv;
        F[i][j] = f;
    }
    __syncthreads();

    // ---------------- blocked Cholesky (4 x 16-wide panels) ----------------
    const int tr = tid >> 4, tc = tid & 15;   // 16x16 thread grid
    for (int k = 0; k < 4; ++k) {
        const int o = k*16;
        // --- factor diagonal tile (lower triangle only is read/used)
        for (int t = 0; t < 16; ++t) {
            if (tid == 0) F[o+t][o+t] = sqrtf(F[o+t][o+t]);
            __syncthreads();
            if (tc == t && tr > t) F[o+tr][o+t] /= F[o+t][o+t];
            __syncthreads();
            if (tr > t && tc > t && tc <= tr)
                F[o+tr][o+tc] -= F[o+tr][o+t]*F[o+tc][o+t];
            __syncthreads();
        }
        // --- TRSM: panel rows below solve  X * Lkk^T = B  (one thread per row)
        const int nrows = (3-k)*16;
        if (tid < nrows) {
            const int r = o + 16 + tid;
            float x[16];
            for (int c = 0; c < 16; ++c) {
                float s = F[r][o+c];
                for (int j2 = 0; j2 < c; ++j2) s -= x[j2]*F[o+c][o+j2];
                x[c] = s / F[o+c][o+c];
                F[r][o+c] = x[c];
            }
        }
        __syncthreads();
        // --- trailing update via WMMA: F[I][J]tile -= Pan_I * Pan_J^T
        const int ntiles = ((3-k)*(4-k)) >> 1;
        const int wave = tid >> 5;            // wave-uniform
        if (wave < ntiles) {
            int I = 0, J = 0, cnt = 0;
            for (int i2 = k+1; i2 < 4; ++i2)
                for (int j2 = k+1; j2 <= i2; ++j2)
                    if (cnt++ == wave) { I = i2; J = j2; }
            const int lane = tid & 31;
            const int half = lane >> 4;
            const int r16  = lane & 15;
            v8f c;
            for (int v = 0; v < 8; ++v)
                c[v] = F[I*16 + v + 8*half][J*16 + r16];
            for (int kk = 0; kk < 16; kk += 4) {
                const int col = o + kk + 2*half;
                v2f a, bf;
                a.x  = -F[I*16 + r16][col];      // negate A => D = C - A*B
                a.y  = -F[I*16 + r16][col+1];
                bf.x =  F[J*16 + r16][col];
                bf.y =  F[J*16 + r16][col+1];
                c = __builtin_amdgcn_wmma_f32_16x16x4_f32(
                        false, a, false, bf, (short)0, c, false, false);
            }
            for (int v = 0; v < 8; ++v)
                F[I*16 + v + 8*half][J*16 + r16] = c[v];
        }
        __syncthreads();
    }

    // ---------------- two RHS: b1 = -chi (valid), b2 = valid mask ----------------
    if (tid < 2*MM) {
        const int s = tid >> 6, i = tid & 63;
        rhs[s][i] = (i < n) ? (s == 0 ? -eln[i] : 1.0f) : 0.0f;
    }
    __syncthreads();

    // forward solve L z = b (column sweep)
    {
        const int s = tid >> 6, r = tid & 63;
        for (int c = 0; c < MM; ++c) {
            if (tid == 0)  rhs[0][c] /= F[c][c];
            if (tid == 64) rhs[1][c] /= F[c][c];
            __syncthreads();
            if (s < 2 && r > c) rhs[s][r] -= F[r][c]*rhs[s][c];
            __syncthreads();
        }
        // backward solve L^T y = z
        for (int c = MM-1; c >= 0; --c) {
            if (tid == 0)  rhs[0][c] /= F[c][c];
            if (tid == 64) rhs[1][c] /= F[c][c];
            __syncthreads();
            if (s < 2 && r < c) rhs[s][r] -= F[c][r]*rhs[s][c];
            __syncthreads();
        }
    }

    // ---------------- Lagrange multiplier, charges ----------------
    if (tid == 0) {
        float s1 = 0.0f, s2 = 0.0f;
        for (int i = 0; i < MM; ++i) { s1 += rhs[0][i]; s2 += rhs[1][i]; }
        lambda_s = (s1 - total_charge[b]) / s2;
    }
    __syncthreads();
    const float lam = lambda_s;
    if (tid < MM) {
        float q = rhs[0][tid] - lam*rhs[1][tid];
        if (tid >= n) q = 0.0f;
        rhs[0][tid] = q;
        out_charge[b*MM + tid] = q;
    }
    __syncthreads();

    // ---------------- energy: e_self + 0.5 * q^T P q ----------------
    float part = 0.0f;
    for (int e = tid; e < MM*MM; e += 256) {
        const int i = e >> 6, j = e & 63;
        part += P[i][j]*rhs[0][i]*rhs[0][j];
    }
    part *= 0.5f;
    if (tid < MM && tid < n) {
        const float q = rhs[0][tid];
        part += 0.5f/(sig[tid]*SQRTPI)*BOHR * q * q;
    }
    red[tid] = part;
    __syncthreads();
    for (int sft = 128; sft > 0; sft >>= 1) {
        if (tid < sft) red[tid] += red[tid + sft];
        __syncthreads();
    }
    if (tid == 0) out_energy[b] = red[0];
}

extern "C" void kernel_launch(void* const* d_in, const int* in_sizes, int n_in,
                              void* d_out, int out_size, void* d_ws, size_t ws_size,
                              hipStream_t stream) {
    const int*   index     = (const int*)  d_in[0];
    const float* positions = (const float*)d_in[1];
    const float* hardness  = (const float*)d_in[2];
    const float* chi       = (const float*)d_in[3];
    const float* sigma     = (const float*)d_in[4];
    const float* totq      = (const float*)d_in[5];
    const int B = in_sizes[0] / MM;           // index is [B, 64]
    float* out_charge = (float*)d_out;        // [B, 64]
    float* out_energy = out_charge + (size_t)B*MM;  // [B]
    qeq_fused_kernel<<<B, 256, 0, stream>>>(index, positions, hardness, chi,
                                            sigma, totq, out_charge, out_energy);
}